// ExpertsFeedForward_64012192580034
// MI455X (gfx1250) — compile-verified
//
#include <hip/hip_runtime.h>
#include <hip/hip_bf16.h>
#include <math.h>

typedef __attribute__((ext_vector_type(2))) float v2f;
typedef __attribute__((ext_vector_type(8))) float v8f;
typedef __attribute__((ext_vector_type(2))) int v2i;
typedef __attribute__((ext_vector_type(4))) int v4i;

typedef __attribute__((address_space(1))) v2i g_v2i;  // global int2
typedef __attribute__((address_space(1))) v4i g_v4i;  // global int4
typedef __attribute__((address_space(3))) v2i s_v2i;  // LDS int2
typedef __attribute__((address_space(3))) v4i s_v4i;  // LDS int4

#define D_MODEL 1024
#define HIDDEN  4096
#define NUM_EXPERTS 8
#define TOKENS  4096

#define BM 128
#define BN 128
#define BK 16
#define LDA 18    // A tile LDS row stride (floats): odd*2 -> conflict-free b64 frags, 8B aligned
#define LDB2 136  // B tile LDS row stride (floats): 136 mod 32 = 8 -> K-halves on disjoint banks; 16B aligned

// Async global->LDS staging if the toolchain exposes the gfx1250 builtins.
#if defined(__has_builtin)
#if __has_builtin(__builtin_amdgcn_global_load_async_to_lds_b64) && \
    __has_builtin(__builtin_amdgcn_global_load_async_to_lds_b128)
#define USE_ASYNC_LDS 1
#endif
#endif

__device__ __forceinline__ void wait_async_done() {
#if defined(USE_ASYNC_LDS)
#if __has_builtin(__builtin_amdgcn_s_wait_asynccnt)
    __builtin_amdgcn_s_wait_asynccnt(0);
#else
    asm volatile("s_wait_asynccnt 0x0" ::: "memory");
#endif
#endif
}

__device__ __forceinline__ float gelu_tanh(float x) {
    const float c = 0.7978845608028654f;  // sqrt(2/pi)
    return 0.5f * x * (1.0f + tanhf(c * (x + 0.044715f * x * x * x)));
}

// ---------------------------------------------------------------------------
// Router: logits = x @ Wg  -> top-2 -> softmax(2) -> dense combine [T, E]
// ---------------------------------------------------------------------------
__global__ __launch_bounds__(256) void moe_router_kernel(
    const float* __restrict__ x, const float* __restrict__ Wg,
    float* __restrict__ combine, float* __restrict__ tail, int tail_n) {
    __shared__ float wg_s[D_MODEL * NUM_EXPERTS];   // 32 KB
    for (int i = threadIdx.x; i < D_MODEL * NUM_EXPERTS; i += 256)
        wg_s[i] = Wg[i];
    __syncthreads();

    const int t = blockIdx.x * 256 + threadIdx.x;
    const float* xt = x + (size_t)t * D_MODEL;

    float lg[NUM_EXPERTS];
#pragma unroll
    for (int e = 0; e < NUM_EXPERTS; ++e) lg[e] = 0.0f;

    for (int k = 0; k < D_MODEL; k += 4) {
        float4 xv = *(const float4*)(xt + k);
        float xs[4] = {xv.x, xv.y, xv.z, xv.w};
#pragma unroll
        for (int j = 0; j < 4; ++j) {
            const float* w = &wg_s[(k + j) * NUM_EXPERTS];
#pragma unroll
            for (int e = 0; e < NUM_EXPERTS; ++e)
                lg[e] = fmaf(xs[j], w[e], lg[e]);
        }
    }

    // top-2, first occurrence wins ties (matches jax.lax.top_k ordering)
    int i1 = 0; float s1 = lg[0];
#pragma unroll
    for (int e = 1; e < NUM_EXPERTS; ++e)
        if (lg[e] > s1) { s1 = lg[e]; i1 = e; }
    int i2 = (i1 == 0) ? 1 : 0; float s2 = lg[i2];
#pragma unroll
    for (int e = 0; e < NUM_EXPERTS; ++e)
        if (e != i1 && lg[e] > s2) { s2 = lg[e]; i2 = e; }

    // softmax over [s1, s2], s1 >= s2
    float w1 = 1.0f / (1.0f + expf(s2 - s1));
    float w2 = 1.0f - w1;

    float* ct = combine + (size_t)t * NUM_EXPERTS;
#pragma unroll
    for (int e = 0; e < NUM_EXPERTS; ++e) ct[e] = 0.0f;
    ct[i1] = w1;
    ct[i2] = w2;

    // router_loss (training=False) -> 0.0 in trailing output slot(s)
    if (blockIdx.x == 0 && threadIdx.x < tail_n) tail[threadIdx.x] = 0.0f;
}

// ---------------------------------------------------------------------------
// Stage one 128x16 A tile and one 16x128 B tile into LDS.
// Async path: global_load_async_to_lds_b64/b128, no VGPR round-trip.
// ---------------------------------------------------------------------------
__device__ __forceinline__ void stage_tile(
    const float* __restrict__ A, const float* __restrict__ B,
    float* Ad, float* Bd, int bm, int bn, int k0, int K, int N, int tid) {
#if defined(USE_ASYNC_LDS)
#pragma unroll
    for (int i = 0; i < 4; ++i) {            // A: 1024 b64 chunks
        int idx = tid + i * 256;             // 0..1023
        int r   = idx >> 3;                  // 0..127
        int c2  = (idx & 7) << 1;            // 0,2,..,14
        __builtin_amdgcn_global_load_async_to_lds_b64(
            (g_v2i*)(A + (size_t)(bm + r) * K + (k0 + c2)),
            (s_v2i*)(Ad + r * LDA + c2), 0, 0);
    }
#pragma unroll
    for (int i = 0; i < 2; ++i) {            // B: 512 b128 chunks
        int idx = tid + i * 256;             // 0..511
        int kk  = idx >> 5;                  // 0..15
        int c4  = (idx & 31) << 2;           // 0,4,..,124
        __builtin_amdgcn_global_load_async_to_lds_b128(
            (g_v4i*)(B + (size_t)(k0 + kk) * N + (bn + c4)),
            (s_v4i*)(Bd + kk * LDB2 + c4), 0, 0);
    }
#else
#pragma unroll
    for (int i = 0; i < 2; ++i) {            // A: 512 float4
        int idx = tid + i * 256;
        int r   = idx >> 2;
        int c4  = (idx & 3) << 2;
        float4 v = *(const float4*)(A + (size_t)(bm + r) * K + (k0 + c4));
        v2f lo = {v.x, v.y};
        v2f hi = {v.z, v.w};
        v2f* p = (v2f*)&Ad[r * LDA + c4];
        p[0] = lo;
        p[1] = hi;
    }
#pragma unroll
    for (int i = 0; i < 2; ++i) {            // B: 512 float4
        int idx = tid + i * 256;
        int kk  = idx >> 5;
        int c4  = (idx & 31) << 2;
        *(float4*)&Bd[kk * LDB2 + c4] =
            *(const float4*)(B + (size_t)(k0 + kk) * N + (bn + c4));
    }
#endif
}

// ---------------------------------------------------------------------------
// Tiled f32 WMMA GEMM: C_tile <- op(A[M,K] @ B[K,N] + bias[N])
//   mode 0: C = gelu(A@B + bias)                      (GEMM1 -> h)
//   mode 1: C = A@B + bias                            (shared expert, write)
//   mode 2: C += combine[row*E+expert] * (A@B + bias) (routed, accumulate)
// Block: 256 threads = 8 waves, tile 128x128, BK=16, double-buffered LDS.
// Wave grid 2x4; each wave: 64x32 via 4x2 v8f accumulators.
// ---------------------------------------------------------------------------
__global__ __launch_bounds__(256) void moe_gemm_wmma(
    const float* __restrict__ A, const float* __restrict__ B,
    const float* __restrict__ bias, float* __restrict__ C,
    const float* __restrict__ combine, int M, int N, int K,
    int expert, int mode) {
    __shared__ float As[2][BM * LDA];    // 2 x 9 KB
    __shared__ float Bs[2][BK * LDB2];   // 2 x 8.5 KB

    const int tid  = threadIdx.x;
    const int lane = tid & 31;
    const int wave = tid >> 5;
    const int wm   = wave >> 2;      // 0..1 : 64-row slab
    const int wn   = wave & 3;       // 0..3 : 32-col slab
    const int half = lane >> 4;      // 0/1  : K-halves of WMMA fragments
    const int lrow = lane & 15;      // row (A frag) / col (B,C frag) in tile

    const int bm = blockIdx.y * BM;
    const int bn = blockIdx.x * BN;

    v8f acc[4][2];
    const v8f vzero = {0.f, 0.f, 0.f, 0.f, 0.f, 0.f, 0.f, 0.f};
#pragma unroll
    for (int ms = 0; ms < 4; ++ms)
#pragma unroll
        for (int ns = 0; ns < 2; ++ns) acc[ms][ns] = vzero;

    // Prime the pipeline: tile 0 -> buffer 0.
    stage_tile(A, B, &As[0][0], &Bs[0][0], bm, bn, 0, K, N, tid);

    for (int k0 = 0, it = 0; k0 < K; k0 += BK, ++it) {
        const int buf = it & 1;
        wait_async_done();           // this wave's async stores to LDS landed
        __syncthreads();             // all waves' staging of `buf` complete

        if (k0 + BK < K)             // stage next tile into the other buffer
            stage_tile(A, B, &As[buf ^ 1][0], &Bs[buf ^ 1][0],
                       bm, bn, k0 + BK, K, N, tid);

        const float* Ab = &As[buf][0];
        const float* Bb = &Bs[buf][0];

#pragma unroll
        for (int ks = 0; ks < BK; ks += 4) {
            const int kb = ks + 2 * half;
            v2f a[4], b[2];
#pragma unroll
            for (int ms = 0; ms < 4; ++ms) {
                int r = wm * 64 + ms * 16 + lrow;
                a[ms] = *(const v2f*)&Ab[r * LDA + kb];
            }
#pragma unroll
            for (int ns = 0; ns < 2; ++ns) {
                int c = wn * 32 + ns * 16 + lrow;
                v2f t;
                t.x = Bb[kb * LDB2 + c];
                t.y = Bb[(kb + 1) * LDB2 + c];
                b[ns] = t;
            }
#pragma unroll
            for (int ms = 0; ms < 4; ++ms)
#pragma unroll
                for (int ns = 0; ns < 2; ++ns)
                    acc[ms][ns] = __builtin_amdgcn_wmma_f32_16x16x4_f32(
                        false, a[ms], false, b[ns], (short)0, acc[ms][ns],
                        false, false);
        }
        // No trailing barrier: next iteration's wait+barrier protects reuse.
    }

    // ---- epilogue: C/D layout is VGPR v -> M = v + 8*half, N = lrow ----
#pragma unroll
    for (int ms = 0; ms < 4; ++ms) {
#pragma unroll
        for (int v = 0; v < 8; ++v) {
            int row = bm + wm * 64 + ms * 16 + v + 8 * half;
            float scale = 1.0f;
            if (mode == 2) scale = combine[(size_t)row * NUM_EXPERTS + expert];
#pragma unroll
            for (int ns = 0; ns < 2; ++ns) {
                int col = bn + wn * 32 + ns * 16 + lrow;
                float val = acc[ms][ns][v] + bias[col];
                float* cp = C + (size_t)row * N + col;
                if (mode == 0)      *cp = gelu_tanh(val);
                else if (mode == 1) *cp = val;
                else                *cp += scale * val;
            }
        }
    }
}

// ---------------------------------------------------------------------------
extern "C" void kernel_launch(void* const* d_in, const int* in_sizes, int n_in,
                              void* d_out, int out_size, void* d_ws, size_t ws_size,
                              hipStream_t stream) {
    const float* x   = (const float*)d_in[0];  // [2,2048,1024]
    const float* Wg  = (const float*)d_in[1];  // [1024,8]
    const float* Wk  = (const float*)d_in[2];  // [8,1024,4096]
    const float* bk  = (const float*)d_in[3];  // [8,4096]
    const float* Wv  = (const float*)d_in[4];  // [8,4096,1024]
    const float* bv  = (const float*)d_in[5];  // [8,1024]
    const float* Wsk = (const float*)d_in[6];  // [1,1024,4096]
    const float* bsk = (const float*)d_in[7];  // [1,4096]
    const float* Wsv = (const float*)d_in[8];  // [1,4096,1024]
    const float* bsv = (const float*)d_in[9];  // [1,1024]
    float* out = (float*)d_out;

    // workspace: combine [T,E] then h [T,HIDDEN]
    float* combine = (float*)d_ws;
    float* h = (float*)((char*)d_ws + (size_t)TOKENS * NUM_EXPERTS * sizeof(float));

    int tail = out_size - TOKENS * D_MODEL;   // router_loss slot(s)
    if (tail < 0) tail = 0;

    moe_router_kernel<<<TOKENS / 256, 256, 0, stream>>>(
        x, Wg, combine, out + (size_t)TOKENS * D_MODEL, tail);

    dim3 block(256);
    dim3 g1(HIDDEN / BN, TOKENS / BM);    // 32 x 32 blocks
    dim3 g2(D_MODEL / BN, TOKENS / BM);   //  8 x 32 blocks

    // Shared expert first: writes out (S==1 so the /S average is identity).
    moe_gemm_wmma<<<g1, block, 0, stream>>>(x, Wsk, bsk, h, nullptr,
                                            TOKENS, HIDDEN, D_MODEL, 0, 0);
    moe_gemm_wmma<<<g2, block, 0, stream>>>(h, Wsv, bsv, out, nullptr,
                                            TOKENS, D_MODEL, HIDDEN, 0, 1);

    // Routed experts: dense compute, weighted accumulate (training path).
    for (int e = 0; e < NUM_EXPERTS; ++e) {
        moe_gemm_wmma<<<g1, block, 0, stream>>>(
            x, Wk + (size_t)e * D_MODEL * HIDDEN, bk + (size_t)e * HIDDEN,
            h, nullptr, TOKENS, HIDDEN, D_MODEL, e, 0);
        moe_gemm_wmma<<<g2, block, 0, stream>>>(
            h, Wv + (size_t)e * HIDDEN * D_MODEL, bv + (size_t)e * D_MODEL,
            out, combine, TOKENS, D_MODEL, HIDDEN, e, 2);
    }
}